// GradientConsistencyLoss_84378927497559
// MI455X (gfx1250) — compile-verified
//
#include <hip/hip_runtime.h>

// Problem geometry: f[B=2][C=3][D=192][H=192][W=192], float32.
static constexpr int  WW = 192;
static constexpr int  SH = 192;                   // stride for h (elements)
static constexpr int  SD = 192 * 192;             // stride for d
static constexpr long long SC = 192LL * 192 * 192;     // stride for channel
static constexpr long long SB = 3LL * SC;              // stride for batch
static constexpr double NVOX = 2.0 * 192.0 * 192.0 * 192.0;

typedef unsigned int u32x4 __attribute__((ext_vector_type(4)));
typedef int          i32x4 __attribute__((ext_vector_type(4)));
typedef int          i32x8 __attribute__((ext_vector_type(8)));

#if defined(__gfx1250__) && __has_builtin(__builtin_amdgcn_tensor_load_to_lds)
#define USE_TDM 1
#endif

#ifdef USE_TDM
// Issue one TDM load of a 2D tile: tile_dim0 = 192 floats (w), tile_dim1 = 3
// rows (channels), row stride = SC elements. Descriptor per CDNA5 ISA §8.
__device__ __forceinline__ void tdm_tile_load(unsigned lds_addr,
                                              unsigned long long gaddr) {
  u32x4 g0;
  g0[0] = 1u;                                          // count=1 (valid), user mode
  g0[1] = lds_addr;                                    // lds_addr (bytes)
  g0[2] = (unsigned)(gaddr & 0xFFFFFFFFull);           // global_addr[31:0]
  g0[3] = (unsigned)((gaddr >> 32) & 0x01FFFFFFull)    // global_addr[56:32]
        | (2u << 30);                                  // type = 2 ("image")
  i32x8 g1;
  g1[0] = (int)(2u << 16);                 // wg_mask=0, data_size=2 (4 bytes)
  g1[1] = (int)(192u << 16);               // tensor_dim0[15:0]=192 (bits 63:48)
  g1[2] = (int)(3u << 16);                 // tensor_dim0 hi=0 | tensor_dim1=3
  g1[3] = (int)(192u << 16);               // tensor_dim1 hi=0 | tile_dim0=192
  g1[4] = (int)(3u);                       // tile_dim1=3, tile_dim2=0
  g1[5] = (int)(unsigned)(SC & 0xFFFFFFFFll); // tensor_dim0_stride[31:0]
  g1[6] = 0;                               // stride[47:32]=0, dim1_stride lo=0
  g1[7] = 0;                               // dim1_stride hi=0
  i32x4 z4 = {0, 0, 0, 0};
#if __has_include(<hip/amd_detail/amd_gfx1250_TDM.h>)
  // therock/clang-23 lane: 6-arg builtin
  i32x8 z8 = {0, 0, 0, 0, 0, 0, 0, 0};
  __builtin_amdgcn_tensor_load_to_lds(g0, g1, z4, z4, z8, 0);
#else
  // ROCm 7.2 / clang-22 lane: 5-arg builtin
  __builtin_amdgcn_tensor_load_to_lds(g0, g1, z4, z4, 0);
#endif
}
#endif

__global__ void gcl_zero(double* acc) {
  acc[0] = 0.0;
  acc[1] = 0.0;
}

__global__ __launch_bounds__(192)
void gcl_main(const float* __restrict__ f, double* __restrict__ acc) {
  // 5 tiles x 3 channels x 192 floats = 11.25 KB staging buffer.
  __shared__ float T[15 * 192];
  __shared__ float wpart[2][8];

  const int w = threadIdx.x;          // 0..191
  int blk = blockIdx.x;
  const int h = blk % 192; blk /= 192;
  const int d = blk % 192; blk /= 192;
  const int b = blk;                  // 0..1

  // jnp.gradient: clamped neighbors + 0.5 (central) / 1.0 (one-sided) scale.
  const int dm = (d > 0) ? d - 1 : 0;
  const int dp = (d < 191) ? d + 1 : 191;
  const int hm = (h > 0) ? h - 1 : 0;
  const int hp = (h < 191) ? h + 1 : 191;
  const float facD = (dp - dm == 2) ? 0.5f : 1.0f;
  const float facH = (hp - hm == 2) ? 0.5f : 1.0f;

  const long long base = (long long)b * SB;
  long long rowoff[5];
  rowoff[0] = base + (long long)d  * SD + (long long)h  * SH;  // center
  rowoff[1] = base + (long long)dm * SD + (long long)h  * SH;  // d-1
  rowoff[2] = base + (long long)dp * SD + (long long)h  * SH;  // d+1
  rowoff[3] = base + (long long)d  * SD + (long long)hm * SH;  // h-1
  rowoff[4] = base + (long long)d  * SD + (long long)hp * SH;  // h+1

#ifdef USE_TDM
  if (threadIdx.x < 32) {   // wave 0 drives the Tensor Data Mover
    // Low 32 bits of a flat LDS pointer == LDS byte offset (ISA 10.2).
    const unsigned lds_base = (unsigned)(unsigned long long)(void*)&T[0];
#pragma unroll
    for (int t = 0; t < 5; ++t) {
      tdm_tile_load(lds_base + (unsigned)(t * 3 * 192 * 4),
                    (unsigned long long)(const void*)(f + rowoff[t]));
    }
    __builtin_amdgcn_s_wait_tensorcnt(0);
  }
  __syncthreads();
#else
  // Fallback staging path (also used by the host-side parse).
#pragma unroll
  for (int r = 0; r < 15; ++r) {
    const int t = r / 3, c = r % 3;
    T[r * 192 + w] = f[rowoff[t] + (long long)c * SC + w];
  }
  __syncthreads();
#endif

  const int wm = (w > 0) ? w - 1 : 0;
  const int wp = (w < 191) ? w + 1 : 191;
  const float facW = (wp - wm == 2) ? 0.5f : 1.0f;

#define TT(t, c, x) T[((t) * 3 + (c)) * 192 + (x)]
  const float gx0 = (TT(2, 0, w) - TT(1, 0, w)) * facD;
  const float gx1 = (TT(2, 1, w) - TT(1, 1, w)) * facD;
  const float gx2 = (TT(2, 2, w) - TT(1, 2, w)) * facD;
  const float gy0 = (TT(4, 0, w) - TT(3, 0, w)) * facH;
  const float gy1 = (TT(4, 1, w) - TT(3, 1, w)) * facH;
  const float gy2 = (TT(4, 2, w) - TT(3, 2, w)) * facH;
  const float gz0 = (TT(0, 0, wp) - TT(0, 0, wm)) * facW;
  const float gz1 = (TT(0, 1, wp) - TT(0, 1, wm)) * facW;
  const float gz2 = (TT(0, 2, wp) - TT(0, 2, wm)) * facW;
#undef TT

  const float dv = gx0 + gy1 + gz2;
  float sdiv = dv * dv;
  const float cx = gz1 - gy2;
  const float cy = gx2 - gz0;
  const float cz = gy0 - gx1;
  float scurl = sqrtf(cx * cx + cy * cy + cz * cz + 1e-8f);

  // wave32 tree reduction
#pragma unroll
  for (int off = 16; off > 0; off >>= 1) {
    sdiv  += __shfl_down(sdiv,  off, 32);
    scurl += __shfl_down(scurl, off, 32);
  }
  const int lane = threadIdx.x & 31;
  const int wid  = threadIdx.x >> 5;  // 0..5
  if (lane == 0) { wpart[0][wid] = sdiv; wpart[1][wid] = scurl; }
  __syncthreads();
  if (threadIdx.x == 0) {
    double a = 0.0, c = 0.0;
#pragma unroll
    for (int i = 0; i < 6; ++i) {
      a += (double)wpart[0][i];
      c += (double)wpart[1][i];
    }
    atomicAdd(&acc[0], a);
    atomicAdd(&acc[1], c);
  }
}

__global__ void gcl_final(const double* __restrict__ acc,
                          float* __restrict__ out) {
  out[0] = (float)(0.1 * (acc[0] / NVOX + acc[1] / NVOX));
}

extern "C" void kernel_launch(void* const* d_in, const int* in_sizes, int n_in,
                              void* d_out, int out_size, void* d_ws, size_t ws_size,
                              hipStream_t stream) {
  const float* f = (const float*)d_in[0];
  double* acc = (double*)d_ws;           // 2 doubles of scratch
  gcl_zero<<<1, 1, 0, stream>>>(acc);
  gcl_main<<<dim3(2 * 192 * 192), dim3(192), 0, stream>>>(f, acc);
  gcl_final<<<1, 1, 0, stream>>>(acc, (float*)d_out);
}